// SO8TGroupStructure_25838523252905
// MI455X (gfx1250) — compile-verified
//
#include <hip/hip_runtime.h>
#include <hip/hip_bf16.h>

typedef __attribute__((ext_vector_type(2))) float v2f;
typedef __attribute__((ext_vector_type(8))) float v8f;

#define ROT 8

// ---------------------------------------------------------------------------
// Setup: build combined rotation Rc = R_safe @ R_cmd * (w_safe*w_cmd).
// Tiny (8x8) — one thread does everything; cost is noise vs the 256 MiB stream.
// ---------------------------------------------------------------------------
__device__ static float det8_inplace(float M[ROT][ROT]) {
    float det = 1.0f;
    for (int i = 0; i < ROT; ++i) {
        int p = i;
        float mx = fabsf(M[i][i]);
        for (int r = i + 1; r < ROT; ++r) {
            float a = fabsf(M[r][i]);
            if (a > mx) { mx = a; p = r; }
        }
        if (p != i) {
            for (int c = 0; c < ROT; ++c) {
                float t = M[i][c]; M[i][c] = M[p][c]; M[p][c] = t;
            }
            det = -det;
        }
        det *= M[i][i];
        float inv = 1.0f / M[i][i];
        for (int r = i + 1; r < ROT; ++r) {
            float f = M[r][i] * inv;
            for (int c = i; c < ROT; ++c) M[r][c] -= f * M[i][c];
        }
    }
    return det;
}

__device__ static void rotation_matrix(const float* __restrict__ p, float R[ROT][ROT]) {
    // row-wise L2 normalize
    for (int i = 0; i < ROT; ++i) {
        float s = 0.0f;
        for (int j = 0; j < ROT; ++j) s += p[i * ROT + j] * p[i * ROT + j];
        float inv = rsqrtf(s);
        for (int j = 0; j < ROT; ++j) R[i][j] = p[i * ROT + j] * inv;
    }
    float M[ROT][ROT];
    for (int i = 0; i < ROT; ++i)
        for (int j = 0; j < ROT; ++j) M[i][j] = R[i][j];
    float det = det8_inplace(M);
    float sign = (det < 0.0f) ? -1.0f : 1.0f;
    float ad = fabsf(det);
    // Qf = Q*sign / |det|^(1/8)
    float scale = sign * powf(ad, -1.0f / (float)ROT);
    for (int i = 0; i < ROT; ++i)
        for (int j = 0; j < ROT; ++j) R[i][j] *= scale;
}

__global__ void so8_setup_kernel(const float* __restrict__ params_safe,
                                 const float* __restrict__ params_cmd,
                                 const float* __restrict__ w_safe,
                                 const float* __restrict__ w_cmd,
                                 float* __restrict__ Rc /* 64 floats */) {
    if (threadIdx.x != 0 || blockIdx.x != 0) return;
    float Rs[ROT][ROT], Rm[ROT][ROT];
    rotation_matrix(params_safe, Rs);
    rotation_matrix(params_cmd, Rm);
    float s = w_safe[0] * w_cmd[0];
    for (int k = 0; k < ROT; ++k) {
        for (int j = 0; j < ROT; ++j) {
            float acc = 0.0f;
            for (int m = 0; m < ROT; ++m) acc += Rs[k][m] * Rm[m][j];
            Rc[k * ROT + j] = s * acc;
        }
    }
}

// ---------------------------------------------------------------------------
// Main streaming kernel: one 16x16 f32 tile (256 consecutive floats = 32
// SO(8) groups) per wave per iteration, computed as a chain of four
// v_wmma_f32_16x16x4_f32 against Bbig = blockdiag(Rc, Rc).
//
// A 16x4 layout:  lane L holds A[M=L&15][K = (L<16?0:2)] in v0, K+1 in v1.
// B 4x16 layout:  VGPR0 = row K (lanes 0-15) / row K+2 (lanes 16-31), VGPR1 = K+1/K+3.
// D 16x16 layout: VGPR r = row r (lanes 0-15) / row r+8 (lanes 16-31), col = lane&15.
// ---------------------------------------------------------------------------
__global__ void __launch_bounds__(256)
so8_wmma_kernel(const float* __restrict__ x,
                const float* __restrict__ Rc,
                float* __restrict__ out,
                int n_tiles) {
    const int lane   = threadIdx.x & 31;
    const int waveIB = threadIdx.x >> 5;
    const int wavesPerBlock = blockDim.x >> 5;

    const int  n    = lane & 15;             // column within tile
    const int  koff = (lane < 16) ? 0 : 2;   // A/B K sub-offset for upper half-wave
    const int  row  = lane & 15;             // A row
    const int  half = (lane < 16) ? 0 : 8;   // D row offset

    // Loop-invariant B fragments: Bbig[k][n] = blockdiag(Rc,Rc)
    v2f b[4];
#pragma unroll
    for (int kc = 0; kc < 4; ++kc) {
        int k0 = 4 * kc + koff;
        int k1 = k0 + 1;
        float b0 = ((k0 < 8) == (n < 8)) ? Rc[(k0 & 7) * 8 + (n & 7)] : 0.0f;
        float b1 = ((k1 < 8) == (n < 8)) ? Rc[(k1 & 7) * 8 + (n & 7)] : 0.0f;
        b[kc].x = b0;
        b[kc].y = b1;
    }

    int tile = blockIdx.x * wavesPerBlock + waveIB;
    const int stride = (int)gridDim.x * wavesPerBlock;

    for (; tile < n_tiles; tile += stride) {
        const long base = (long)tile * 256;

        v8f c = {};
#pragma unroll
        for (int kc = 0; kc < 4; ++kc) {
            // 8-byte aligned: base%256==0, 16*row%16==0, 4*kc%4==0, koff even
            const v2f a = *reinterpret_cast<const v2f*>(x + base + 16 * row + 4 * kc + koff);
            c = __builtin_amdgcn_wmma_f32_16x16x4_f32(
                    /*neg_a=*/false, a, /*neg_b=*/false, b[kc],
                    /*c_mod=*/(short)0, c, /*reuse_a=*/false, /*reuse_b=*/false);
        }

#pragma unroll
        for (int r = 0; r < 8; ++r) {
            out[base + 16 * (r + half) + n] = c[r];
        }
    }
}

// ---------------------------------------------------------------------------
extern "C" void kernel_launch(void* const* d_in, const int* in_sizes, int n_in,
                              void* d_out, int out_size, void* d_ws, size_t ws_size,
                              hipStream_t stream) {
    const float* x           = (const float*)d_in[0];
    const float* params_safe = (const float*)d_in[1];
    const float* params_cmd  = (const float*)d_in[2];
    const float* w_safe      = (const float*)d_in[3];
    const float* w_cmd       = (const float*)d_in[4];
    float*       out         = (float*)d_out;
    float*       Rc          = (float*)d_ws;   // 64 floats of scratch

    so8_setup_kernel<<<1, 32, 0, stream>>>(params_safe, params_cmd, w_safe, w_cmd, Rc);

    const long n_elems = (long)in_sizes[0];          // 2*4096*4096, multiple of 256
    const int  n_tiles = (int)(n_elems / 256);
    const int  wavesPerBlock = 8;                    // 256 threads, wave32
    int blocks = (n_tiles + wavesPerBlock - 1) / wavesPerBlock;
    if (blocks < 1) blocks = 1;

    so8_wmma_kernel<<<blocks, 256, 0, stream>>>(x, Rc, out, n_tiles);
}